// SelfTensorProductS2Grid_42305427866179
// MI455X (gfx1250) — compile-verified
//
#include <hip/hip_runtime.h>

// Self tensor product on S2 grid, fused:  out = Wfrom^T @ ((Wto @ inp)^2)
// Per n: Wto (324,49) @ inp (49,128) -> grid (324,128); square; Wfrom^T (49,324) @ grid.
// One workgroup per n, 8 waves, each wave owns a 16-wide column strip of C.
// V_WMMA_F32_16X16X4_F32 throughout (full f32; workload is HBM-bound).
// Pre-pass builds zero-padded WtoP and transposed WfT in d_ws so every A fragment
// is one unconditional global_load_b64; LDS strips are column-major so every B
// fragment is one ds_load_b64 into an adjacent register pair.

typedef float v2f __attribute__((ext_vector_type(2)));
typedef float v4f __attribute__((ext_vector_type(4)));
typedef float v8f __attribute__((ext_vector_type(8)));

#define L2C   49            // (lmax+1)^2
#define CCH   128           // channels
#define BA    324           // 18*18 grid points
#define MT1   21            // ceil(324/16) stage-1 M tiles
#define KT1   13            // ceil(49/4)   stage-1 K steps

#define WTO_ROWS 336        // 324 padded to 16
#define WTO_STR  52         // 49 padded to 4 (even -> 8B aligned pairs)
#define WFT_ROWS 64         // 49 padded to 16
#define WFT_STR  336        // 324 padded so ba0+1 (max 335) is in-bounds

#define CSTR1 60            // inp strip col stride (even, conflict-free b64 pattern)
#define TSTR  20            // tile col stride (multiple of 4 for b128 stores)
#define WAVE_LDS (16 * CSTR1 + 16 * TSTR)   // per-wave floats (1280)

// ---------------------------------------------------------------- weight prep
__global__ __launch_bounds__(256)
void prep_weights_kernel(const float* __restrict__ Wto,
                         const float* __restrict__ Wfrom,
                         float* __restrict__ WtoP,     // [336][52]
                         float* __restrict__ WfT)      // [64][336] transposed
{
    const int t = blockIdx.x * 256 + threadIdx.x;
    if (t < WTO_ROWS * WTO_STR) {
        const int r = t / WTO_STR, k = t - r * WTO_STR;
        WtoP[t] = (r < BA && k < L2C) ? Wto[r * L2C + k] : 0.0f;
    }
    if (t < WFT_ROWS * WFT_STR) {
        const int i = t / WFT_STR, ba = t - i * WFT_STR;
        WfT[t] = (i < L2C && ba < BA) ? Wfrom[ba * L2C + i] : 0.0f;
    }
}

// ------------------------------------------------------------------ main pass
template <bool PREP>
__global__ __launch_bounds__(256)
void stp_s2grid_kernel(const float* __restrict__ inp,
                       const float* __restrict__ Wto,
                       const float* __restrict__ Wfrom,
                       const float* __restrict__ WtoP,
                       const float* __restrict__ WfT,
                       float* __restrict__ out)
{
    __shared__ __align__(16) float lds[8 * WAVE_LDS];

    const int n      = blockIdx.x;
    const int tid    = threadIdx.x;
    const int wave   = tid >> 5;
    const int lane   = tid & 31;
    const int lane16 = lane & 15;
    const int khi    = (lane >> 4) << 1;   // 0 (lanes 0-15) or 2 (lanes 16-31)
    const int hi8    = (lane >> 4) << 3;   // 0 or 8 (C/D row offset)

    if (PREP) {
        __builtin_prefetch(WtoP + ((tid * 64) % (WTO_ROWS * WTO_STR)), 0, 0);
        __builtin_prefetch(WfT + ((tid * 64) % (WFT_ROWS * WFT_STR)), 0, 0);
    } else {
        __builtin_prefetch(Wto + ((tid * 64) % (BA * L2C)), 0, 0);
        __builtin_prefetch(Wfrom + ((tid * 64) % (BA * L2C)), 0, 0);
    }

    const float* inpN = inp + (size_t)n * (L2C * CCH);

    // Cooperative coalesced load of inputs[n] into COLUMN-MAJOR per-wave strips:
    // strip layout lds[wave][c][k], c = channel within strip, k = SH coeff.
    for (int idx = tid; idx < L2C * CCH; idx += 256) {
        const int r = idx >> 7;          // SH coeff index (k)
        const int c = idx & 127;         // channel
        lds[(c >> 4) * WAVE_LDS + (c & 15) * CSTR1 + r] = inpN[idx];
    }
    // zero K padding rows (49..51)
    for (int idx = tid; idx < 3 * CCH; idx += 256) {
        const int r = L2C + (idx >> 7);
        const int c = idx & 127;
        lds[(c >> 4) * WAVE_LDS + (c & 15) * CSTR1 + r] = 0.0f;
    }
    __syncthreads();

    float* inpS = &lds[wave * WAVE_LDS];
    float* tile = inpS + 16 * CSTR1;           // [col][row], stride TSTR
    const int colbase = wave << 4;

    v8f outacc[4];
    #pragma unroll
    for (int ot = 0; ot < 4; ++ot) outacc[ot] = (v8f){};

    for (int mt = 0; mt < MT1; ++mt) {
        // ---- stage 1: one 16x16 grid tile = Wto[mt-rows] @ inp_strip --------
        v8f acc = {};
        const int gm = mt * 16 + lane16;                 // ba row (A matrix M)
        for (int kk = 0; kk < KT1; ++kk) {
            const int k0 = kk * 4 + khi;                 // K for VGPR0 of A/B
            v2f a, b;
            if (PREP) {
                a = *(const v2f*)(WtoP + gm * WTO_STR + k0);      // b64, no mask
            } else {
                a.x = (gm < BA && k0     < L2C) ? Wto[gm * L2C + k0    ] : 0.0f;
                a.y = (gm < BA && k0 + 1 < L2C) ? Wto[gm * L2C + k0 + 1] : 0.0f;
            }
            b = *(const v2f*)(inpS + lane16 * CSTR1 + k0);        // ds_load_b64
            acc = __builtin_amdgcn_wmma_f32_16x16x4_f32(
                      false, a, false, b, (short)0, acc, false, false);
        }

        // elementwise self tensor product
        acc = acc * acc;

        // D-layout -> column-major tile: lane owns col lane16, rows hi8..hi8+7
        {
            v4f lo = {acc[0], acc[1], acc[2], acc[3]};
            v4f hi = {acc[4], acc[5], acc[6], acc[7]};
            *(v4f*)(tile + lane16 * TSTR + hi8)     = lo;          // ds_store_b128
            *(v4f*)(tile + lane16 * TSTR + hi8 + 4) = hi;          // ds_store_b128
        }
        asm volatile("s_wait_dscnt 0x0" ::: "memory");

        // ---- stage 2: fold grid tile into the 4 output row-tile accumulators
        #pragma unroll
        for (int kt = 0; kt < 4; ++kt) {
            const int ba0 = (mt * 4 + kt) * 4 + khi;     // stage-2 K (ba index)
            v2f b = *(const v2f*)(tile + lane16 * TSTR + kt * 4 + khi); // b64
            #pragma unroll
            for (int ot = 0; ot < 4; ++ot) {
                const int irow = ot * 16 + lane16;       // SH coeff row (M)
                v2f a;
                if (PREP) {
                    a = *(const v2f*)(WfT + irow * WFT_STR + ba0); // b64, no mask
                } else {
                    a.x = (ba0     < BA && irow < L2C) ? Wfrom[(ba0    ) * L2C + irow] : 0.0f;
                    a.y = (ba0 + 1 < BA && irow < L2C) ? Wfrom[(ba0 + 1) * L2C + irow] : 0.0f;
                }
                outacc[ot] = __builtin_amdgcn_wmma_f32_16x16x4_f32(
                                 false, a, false, b, (short)0, outacc[ot], false, false);
            }
        }
    }

    // ---- write out[n] (only valid SH rows i < 49); 16-lane rows are contiguous
    float* outN = out + (size_t)n * (L2C * CCH);
    #pragma unroll
    for (int ot = 0; ot < 4; ++ot) {
        #pragma unroll
        for (int r = 0; r < 8; ++r) {
            const int i = ot * 16 + r + hi8;
            if (i < L2C) outN[i * CCH + colbase + lane16] = outacc[ot][r];
        }
    }
}

extern "C" void kernel_launch(void* const* d_in, const int* in_sizes, int n_in,
                              void* d_out, int out_size, void* d_ws, size_t ws_size,
                              hipStream_t stream)
{
    const float* inp   = (const float*)d_in[0];
    const float* Wto   = (const float*)d_in[1];
    const float* Wfrom = (const float*)d_in[2];
    float* out = (float*)d_out;

    const int N = in_sizes[0] / (L2C * CCH);   // 4096

    const size_t wtop_elems = (size_t)WTO_ROWS * WTO_STR;   // 17472
    const size_t wft_elems  = (size_t)WFT_ROWS * WFT_STR;   // 21504
    const size_t need_bytes = (wtop_elems + wft_elems) * sizeof(float);

    dim3 block(256);
    if (ws_size >= need_bytes) {
        float* WtoP = (float*)d_ws;
        float* WfT  = WtoP + wtop_elems;
        const int prep_threads = (int)(wft_elems > wtop_elems ? wft_elems : wtop_elems);
        prep_weights_kernel<<<dim3((prep_threads + 255) / 256), block, 0, stream>>>(
            Wto, Wfrom, WtoP, WfT);
        stp_s2grid_kernel<true><<<dim3(N), block, 0, stream>>>(
            inp, Wto, Wfrom, WtoP, WfT, out);
    } else {
        stp_s2grid_kernel<false><<<dim3(N), block, 0, stream>>>(
            inp, Wto, Wfrom, nullptr, nullptr, out);
    }
}